// ANO_VQC_Model_6236292514376
// MI455X (gfx1250) — compile-verified
//
#include <hip/hip_runtime.h>
#include <math.h>

#define NQ      14
#define NSTATE  (1 << NQ)     // 16384 amplitudes
#define DEPTH   6
#define BATCH   512
#define TPB     256           // 8 wave32s
#define NWAVE   (TPB / 32)

typedef float v2f __attribute__((ext_vector_type(2)));
typedef float v8f __attribute__((ext_vector_type(8)));

// LDS layout (bytes):
//   st   : float2[16384] @ 0        (131072)
//   wsum : float[2048]   @ 131072   (8192)   per-wave WMMA partial tiles
//   Hr   : float[64]     @ 139264   (256)
//   Hi   : float[64]     @ 139520   (256)
//   fac  : float[28]     @ 139776   (112)    per-qubit product factors
//   red  : float[64]     @ 139904   (256)
#define SMEM_BYTES 140160

__global__ __launch_bounds__(TPB)
void vqc_kernel(const float* __restrict__ X,     // (512,14)
                const float* __restrict__ theta, // (6,14)
                const float* __restrict__ A,     // (14,28)
                const float* __restrict__ Bm,    // (14,28)
                const float* __restrict__ Dp,    // (14,8)
                float* __restrict__ out)         // (512,2)
{
    extern __shared__ char smem[];
    float2* st   = (float2*)(smem);
    float*  wsum = (float*) (smem + 131072);
    float*  Hr   = (float*) (smem + 139264);
    float*  Hi   = (float*) (smem + 139520);
    float*  fac  = (float*) (smem + 139776);
    float*  red  = (float*) (smem + 139904);

    const int tid = threadIdx.x;
    const int b   = blockIdx.x;

    // ---- per-qubit factors: Hadamard + RY(X[b,w]) on a product state ----
    if (tid < NQ) {
        float x = X[b * NQ + tid];
        float s, c;
        sincosf(0.5f * x, &s, &c);
        const float inv = 0.70710678118654752f;   // 1/sqrt(2) per qubit
        fac[2 * tid + 0] = (c - s) * inv;         // amplitude for bit=0
        fac[2 * tid + 1] = (c + s) * inv;         // amplitude for bit=1
    }
    __syncthreads();

    // ---- initialize state as the 14-fold product (replaces init + 14 RY passes) ----
    for (int idx = tid; idx < NSTATE; idx += TPB) {
        float p = 1.0f;
#pragma unroll
        for (int w = 0; w < NQ; ++w)
            p *= fac[2 * w + ((idx >> (NQ - 1 - w)) & 1)];
        st[idx] = make_float2(p, 0.0f);
    }
    __syncthreads();

    // ---- 6 entangling+rotation layers ----
    for (int k = 0; k < DEPTH; ++k) {
        // even CNOT layer: ctrl qubits 0,2,..,12 (bit positions 13,11,..,1)
        for (int idx = tid; idx < NSTATE; idx += TPB) {
            int p = idx ^ ((idx & 0x2AAA) >> 1);
            if (p > idx) { float2 t = st[idx]; st[idx] = st[p]; st[p] = t; }
        }
        __syncthreads();
        // odd CNOT layer: ctrl qubits 1,3,..,11 (bit positions 12,10,..,2)
        for (int idx = tid; idx < NSTATE; idx += TPB) {
            int p = idx ^ ((idx & 0x1554) >> 1);
            if (p > idx) { float2 t = st[idx]; st[idx] = st[p]; st[p] = t; }
        }
        __syncthreads();

        // RY gates: all 14 act on distinct qubits -> they commute. Fuse adjacent
        // pairs (gp, gp+1) into one 4-amplitude-group pass: halves LDS traffic
        // and barrier count of the dominant cost (7 passes instead of 14).
        for (int gp = 0; gp < NQ; gp += 2) {
            float s1, c1, s2, c2;
            sincosf(0.5f * theta[k * NQ + gp],     &s1, &c1);  // qubit gp  (high bit)
            sincosf(0.5f * theta[k * NQ + gp + 1], &s2, &c2);  // qubit gp+1 (low bit)
            const int p2 = 12 - gp;          // bit position of qubit gp+1
            const int S2 = 1 << p2;
            for (int t = tid; t < NSTATE / 4; t += TPB) {
                int lo  = t & (S2 - 1);
                int i00 = ((t >> p2) << (p2 + 2)) | lo;
                int i01 = i00 | S2;
                int i10 = i00 | (S2 << 1);
                int i11 = i10 | S2;
                float2 a00 = st[i00], a01 = st[i01], a10 = st[i10], a11 = st[i11];
                // rotate low-bit qubit (gp+1)
                float t00x = c2 * a00.x - s2 * a01.x, t00y = c2 * a00.y - s2 * a01.y;
                float t01x = s2 * a00.x + c2 * a01.x, t01y = s2 * a00.y + c2 * a01.y;
                float t10x = c2 * a10.x - s2 * a11.x, t10y = c2 * a10.y - s2 * a11.y;
                float t11x = s2 * a10.x + c2 * a11.x, t11y = s2 * a10.y + c2 * a11.y;
                // rotate high-bit qubit (gp)
                st[i00] = make_float2(c1 * t00x - s1 * t10x, c1 * t00y - s1 * t10y);
                st[i10] = make_float2(s1 * t00x + c1 * t10x, s1 * t00y + c1 * t10y);
                st[i01] = make_float2(c1 * t01x - s1 * t11x, c1 * t01y - s1 * t11y);
                st[i11] = make_float2(s1 * t01x + c1 * t11x, s1 * t01y + c1 * t11y);
            }
            __syncthreads();
        }
    }

    // ---- expectations for q = 0,1 via WMMA Gram matrices ----
    const int  lane  = tid & 31;
    const int  wave  = tid >> 5;
    const int  m     = lane & 15;       // stacked row: 0..7 = Re rows, 8..15 = Im rows
    const int  i8    = m & 7;           // observable index i (3 moved wires)
    const int  imOff = (m >= 8) ? 1 : 0;
    const float* sf  = (const float*)st;

    for (int q = 0; q < 2; ++q) {
        // build 8x8 Hr (symmetric) / Hi (antisymmetric), diag doubled by h + h^H
        if (tid < 64) {
            int i = tid >> 3, j = tid & 7;
            float hr, hi;
            if (i > j)      { int kk = i * (i - 1) / 2 + j; hr = A[q * 28 + kk]; hi =  Bm[q * 28 + kk]; }
            else if (i < j) { int kk = j * (j - 1) / 2 + i; hr = A[q * 28 + kk]; hi = -Bm[q * 28 + kk]; }
            else            { hr = (i < 7) ? 2.0f * Dp[q * 8 + i + 1] : 0.0f; hi = 0.0f; }
            Hr[tid] = hr; Hi[tid] = hi;
        }

        // A-matrix (16 x 2048, rows = [Re psi_i ; Im psi_i]); B = A^T shares the
        // same per-lane register layout, so Gram = wmma(a, a, acc).
        const int shift  = 11 - q;                // bits below the 3 moved wires
        const int maskLo = (1 << shift) - 1;
        const int rbase  = wave * 256 + ((lane >= 16) ? 2 : 0);

        v8f acc = {};
        for (int kk = 0; kk < 64; ++kk) {
            int r0 = rbase + kk * 4;
            int r1 = r0 + 1;
            int f0 = ((r0 >> shift) << (14 - q)) | (i8 << shift) | (r0 & maskLo);
            int f1 = ((r1 >> shift) << (14 - q)) | (i8 << shift) | (r1 & maskLo);
            v2f a;
            a.x = sf[2 * f0 + imOff];
            a.y = sf[2 * f1 + imOff];
            // D(16x16) += A(16x4) * A^T(4x16): top-left Grr, bottom-left Gir, bottom-right Gii
            acc = __builtin_amdgcn_wmma_f32_16x16x4_f32(
                      false, a, false, a, (short)0, acc, false, false);
        }

        // stash per-wave 16x16 partial: element (M,N) -> vgpr M&7, lane (M>=8?16:0)+N
#pragma unroll
        for (int e = 0; e < 8; ++e)
            wsum[wave * 256 + e * 32 + lane] = acc[e];
        __syncthreads();   // wsum + Hr/Hi visible

        // e = <Hr, Grr + Gii> + 2 <Hi, Gir>
        if (tid < 64) {
            int i = tid >> 3, j = tid & 7;
            float rr = 0.0f, ii = 0.0f, ir = 0.0f;
            for (int w = 0; w < NWAVE; ++w) {
                const float* base = wsum + w * 256 + i * 32;
                rr += base[j];          // D[i][j]       = sum Re_i Re_j
                ir += base[16 + j];     // D[i+8][j]     = sum Im_i Re_j
                ii += base[24 + j];     // D[i+8][j+8]   = sum Im_i Im_j
            }
            red[tid] = Hr[i * 8 + j] * (rr + ii) + 2.0f * Hi[i * 8 + j] * ir;
        }
        __syncthreads();
        if (tid == 0) {
            float e = 0.0f;
#pragma unroll
            for (int t = 0; t < 64; ++t) e += red[t];
            out[b * 2 + q] = e;
        }
        __syncthreads();   // guard Hr/Hi/wsum/red reuse for next q
    }
}

extern "C" void kernel_launch(void* const* d_in, const int* in_sizes, int n_in,
                              void* d_out, int out_size, void* d_ws, size_t ws_size,
                              hipStream_t stream) {
    (void)in_sizes; (void)n_in; (void)d_ws; (void)ws_size; (void)out_size;
    const float* X     = (const float*)d_in[0];
    const float* theta = (const float*)d_in[1];
    const float* A     = (const float*)d_in[2];
    const float* Bm    = (const float*)d_in[3];
    const float* Dp    = (const float*)d_in[4];
    float* out = (float*)d_out;

    // allow >64KB dynamic LDS (320KB/WGP on CDNA5); host API, not a stream op
    static bool attr_set = [] {
        hipFuncSetAttribute((const void*)vqc_kernel,
                            hipFuncAttributeMaxDynamicSharedMemorySize, SMEM_BYTES);
        return true;
    }();
    (void)attr_set;

    vqc_kernel<<<dim3(BATCH), dim3(TPB), SMEM_BYTES, stream>>>(X, theta, A, Bm, Dp, out);
}